// MatSurfGcn_85968065397069
// MI455X (gfx1250) — compile-verified
//
#include <hip/hip_runtime.h>
#include <hip/hip_bf16.h>

// ---------------------------------------------------------------------------
// MatSurfGcn on MI455X (gfx1250, wave32).
//
// Roofline: dominated by streaming W_g1 (4096x8192 fp32 = 128MB) exactly once
// -> ~5.8us floor @ 23.3TB/s. ~7 FLOP/byte => memory bound. Keep fp32 (zero
// conversion traffic) and use V_WMMA_F32_16X16X4_F32.
//
// Algebra: gcn(X) = A_hat @ (X @ W) + b = (A_hat @ X) @ W + b, so pre-
// aggregate the tiny [14,4096] activations and do ONE big GEMM:
//   H[16,8192] = X2[16,4096] @ W_g1[4096,8192] + b_g1   (M padded 14->16)
//
// GEMM kernel design (reasoned, compile-only):
//   * W stream uses NONTEMPORAL loads: a 128MB one-shot stream must not
//     evict the 192MB L2's useful contents.
//   * 2 column tiles per wave: A-fragments (shared activations) feed 2 WMMAs
//     each -> half the A-load instructions per MAC.
//   * 2 accumulators per tile (even/odd K-quads): breaks the WMMA C-register
//     RAW chain, 4 independent WMMA streams per wave for the XDL pipe.
//   * K split 8 ways -> 2048 waves for outstanding-load depth; partials are
//     NT-stored and reduced (+bias) in a cheap follow-up kernel.
//   * global_prefetch ~8 iterations ahead on the streamed operand.
// ---------------------------------------------------------------------------

typedef __attribute__((ext_vector_type(2))) float v2f;
typedef __attribute__((ext_vector_type(8))) float v8f;

#define D1 4096
#define D2 8192
#define NNODE 14
#define NPAD 16
#define NEDGE 40
#define KSPLIT 8
#define KCH (D1 / KSPLIT)          // 512
#define TPW 2                      // column tiles per wave
#define NGROUP (D2 / (16 * TPW))   // 256 tile groups
#define NWAVE (NGROUP * KSPLIT)    // 2048 waves

// ---------------- 1) encoders -> X[16][D1], relu, rows 14/15 zero ----------
__global__ void enc_kernel(const float* __restrict__ mats, const float* __restrict__ cyls,
                           const float* __restrict__ planes, const float* __restrict__ power,
                           const float* __restrict__ Wm, const float* __restrict__ bm,
                           const float* __restrict__ Wc, const float* __restrict__ bc,
                           const float* __restrict__ Wp, const float* __restrict__ bp,
                           const float* __restrict__ Ww, const float* __restrict__ bw,
                           float* __restrict__ X)
{
    int idx = blockIdx.x * blockDim.x + threadIdx.x;   // NPAD * D1 threads
    int col = idx & (D1 - 1);
    int row = idx >> 12;                               // D1 == 2^12
    float acc;
    if (row < 6) {
        acc = bm[col];
        #pragma unroll
        for (int f = 0; f < 6; ++f) acc += mats[row * 6 + f] * Wm[f * D1 + col];
    } else if (row < 10) {
        int r = row - 6;
        acc = bc[col];
        #pragma unroll
        for (int f = 0; f < 3; ++f) acc += cyls[r * 3 + f] * Wc[f * D1 + col];
    } else if (row < 13) {
        int r = row - 10;
        acc = bp[col];
        #pragma unroll
        for (int f = 0; f < 4; ++f) acc += planes[r * 4 + f] * Wp[f * D1 + col];
    } else if (row == 13) {
        acc = bw[col] + (power[0] * 1e-4f) * Ww[col];
    } else {
        X[idx] = 0.0f;   // padded rows
        return;
    }
    X[idx] = fmaxf(acc, 0.0f);
}

// ---------------- 2) normalized adjacency with self loops ------------------
__global__ void ahat_kernel(const int* __restrict__ ei, float* __restrict__ Ahat)
{
    if (threadIdx.x != 0 || blockIdx.x != 0) return;
    float deg[NPAD], dis[NPAD];
    for (int i = 0; i < NPAD * NPAD; ++i) Ahat[i] = 0.0f;
    for (int n = 0; n < NPAD; ++n) deg[n] = 0.0f;
    for (int e = 0; e < NEDGE; ++e) deg[ei[NEDGE + e]] += 1.0f;  // dst row
    for (int n = 0; n < NNODE; ++n) deg[n] += 1.0f;              // self loops
    for (int n = 0; n < NPAD; ++n)
        dis[n] = (deg[n] > 0.0f) ? rsqrtf(fmaxf(deg[n], 1e-12f)) : 0.0f;
    for (int e = 0; e < NEDGE; ++e) {
        int s = ei[e], d = ei[NEDGE + e];
        Ahat[d * NPAD + s] += dis[s] * dis[d];
    }
    for (int n = 0; n < NNODE; ++n) Ahat[n * NPAD + n] += dis[n] * dis[n];
}

// ---------------- 3) X2T[k][m] = (A_hat @ X)^T -----------------------------
__global__ void x2t_kernel(const float* __restrict__ X, const float* __restrict__ Ahat,
                           float* __restrict__ X2T)
{
    int idx = blockIdx.x * blockDim.x + threadIdx.x;   // D1 * NPAD
    int m = idx & (NPAD - 1);
    int c = idx >> 4;
    float acc = 0.0f;
    #pragma unroll
    for (int n = 0; n < NPAD; ++n) acc += Ahat[m * NPAD + n] * X[n * D1 + c];
    X2T[idx] = acc;                                    // idx == c*NPAD + m
}

// ---------------- 4) WMMA f32 16x16x4 GEMM ---------------------------------
// A 16x4 layout : lanes 0-15 -> M=0..15 K={0,1}; lanes 16-31 -> K={2,3}
// B 4x16 layout : VGPR0 rows K0 (lanes 0-15) / K2 (lanes 16-31); VGPR1 K1/K3
// C 16x16 layout: VGPR v -> row M=v (lanes 0-15), M=v+8 (lanes 16-31)
__device__ __forceinline__ v8f wmma4(v2f a, v2f b, v8f c)
{
    return __builtin_amdgcn_wmma_f32_16x16x4_f32(
        /*neg_a=*/false, a, /*neg_b=*/false, b,
        /*c_mod=*/(short)0, c, /*reuse_a=*/false, /*reuse_b=*/false);
}

__global__ void gemm_kernel(const float* __restrict__ X2T, const float* __restrict__ W,
                            float* __restrict__ Hp)
{
    int wid  = (blockIdx.x * blockDim.x + threadIdx.x) >> 5;  // 0..NWAVE-1
    int lane = threadIdx.x & 31;
    int grp  = wid & (NGROUP - 1);                            // 32-col group
    int ks   = wid >> 8;                                      // K chunk (NGROUP==256)
    int half = lane >> 4, l16 = lane & 15;
    int n0   = grp * (16 * TPW);

    const float* aptr  = X2T + (size_t)(ks * KCH + 2 * half) * NPAD + l16;
    const float* bptr0 = W   + (size_t)(ks * KCH + 2 * half) * D2 + n0 + l16;
    const float* bptr1 = bptr0 + 16;

    v8f c00 = {0.f,0.f,0.f,0.f,0.f,0.f,0.f,0.f};   // tile0, even K-quads
    v8f c01 = c00;                                  // tile0, odd  K-quads
    v8f c10 = c00;                                  // tile1, even
    v8f c11 = c00;                                  // tile1, odd

    #pragma unroll 4
    for (int k = 0; k < KCH; k += 8) {
        // even K-quad (rows k .. k+3)
        v2f a0, b00, b10;
        a0.x  = aptr[0];                     a0.y  = aptr[NPAD];
        b00.x = __builtin_nontemporal_load(bptr0);
        b00.y = __builtin_nontemporal_load(bptr0 + D2);
        b10.x = __builtin_nontemporal_load(bptr1);
        b10.y = __builtin_nontemporal_load(bptr1 + D2);
        __builtin_prefetch(bptr0 + (size_t)64 * D2, 0, 2);   // ~8 iters ahead
        __builtin_prefetch(bptr1 + (size_t)64 * D2, 0, 2);
        c00 = wmma4(a0, b00, c00);
        c10 = wmma4(a0, b10, c10);
        // odd K-quad (rows k+4 .. k+7)
        v2f a1, b01, b11;
        a1.x  = aptr[4 * NPAD];              a1.y  = aptr[5 * NPAD];
        b01.x = __builtin_nontemporal_load(bptr0 + (size_t)4 * D2);
        b01.y = __builtin_nontemporal_load(bptr0 + (size_t)5 * D2);
        b11.x = __builtin_nontemporal_load(bptr1 + (size_t)4 * D2);
        b11.y = __builtin_nontemporal_load(bptr1 + (size_t)5 * D2);
        c01 = wmma4(a1, b01, c01);
        c11 = wmma4(a1, b11, c11);
        aptr  += 8 * NPAD;
        bptr0 += (size_t)8 * D2;
        bptr1 += (size_t)8 * D2;
    }

    float* out = Hp + (size_t)ks * NPAD * D2;
    #pragma unroll
    for (int v = 0; v < 8; ++v) {
        int row = v + half * 8;
        __builtin_nontemporal_store(c00[v] + c01[v], out + (size_t)row * D2 + n0 + l16);
        __builtin_nontemporal_store(c10[v] + c11[v], out + (size_t)row * D2 + n0 + 16 + l16);
    }
}

// ---------------- 5) sum K-partials + bias -> H ----------------------------
__global__ void reduce_kernel(const float* __restrict__ Hp, const float* __restrict__ bg1,
                              float* __restrict__ H)
{
    int idx = blockIdx.x * blockDim.x + threadIdx.x;   // NPAD * D2
    float s = bg1[idx & (D2 - 1)];
    #pragma unroll
    for (int ks = 0; ks < KSPLIT; ++ks)
        s += __builtin_nontemporal_load(Hp + (size_t)ks * NPAD * D2 + idx);
    H[idx] = s;
}

// ---------------- 6) conv2 + regression head -------------------------------
__global__ void final_kernel(const float* __restrict__ H, const float* __restrict__ Ahat,
                             const float* __restrict__ Wg2, const float* __restrict__ bg2,
                             const float* __restrict__ Whead, const float* __restrict__ bhead,
                             float* __restrict__ out)
{
    __shared__ float t[NPAD];
    __shared__ float red[256];
    int tid = threadIdx.x;
    for (int m = 0; m < NNODE; ++m) {
        float s = 0.0f;
        for (int k = tid; k < D2; k += 256) s += H[(size_t)m * D2 + k] * Wg2[k];
        red[tid] = s;
        __syncthreads();
        for (int off = 128; off > 0; off >>= 1) {
            if (tid < off) red[tid] += red[tid + off];
            __syncthreads();
        }
        if (tid == 0) t[m] = red[0];      // (H @ W_g2)[m]
        __syncthreads();
    }
    if (tid == 0) {
        float y = 0.0f;
        for (int m = 0; m < NNODE; ++m) {
            float x3 = bg2[0];
            for (int n = 0; n < NNODE; ++n) x3 += Ahat[m * NPAD + n] * t[n];
            y += x3 * Whead[m];
        }
        out[0] = y + bhead[0];            // mean over a [1,1] is identity
    }
}

// ---------------------------------------------------------------------------
extern "C" void kernel_launch(void* const* d_in, const int* in_sizes, int n_in,
                              void* d_out, int out_size, void* d_ws, size_t ws_size,
                              hipStream_t stream) {
    (void)in_sizes; (void)n_in; (void)out_size; (void)ws_size;
    const float* mats   = (const float*)d_in[0];
    const float* cyls   = (const float*)d_in[1];
    const float* planes = (const float*)d_in[2];
    const float* power  = (const float*)d_in[3];
    const int*   ei     = (const int*)  d_in[4];
    const float* W_mat  = (const float*)d_in[5];
    const float* b_mat  = (const float*)d_in[6];
    const float* W_cyl  = (const float*)d_in[7];
    const float* b_cyl  = (const float*)d_in[8];
    const float* W_pl   = (const float*)d_in[9];
    const float* b_pl   = (const float*)d_in[10];
    const float* W_pw   = (const float*)d_in[11];
    const float* b_pw   = (const float*)d_in[12];
    const float* W_g1   = (const float*)d_in[13];
    const float* b_g1   = (const float*)d_in[14];
    const float* W_g2   = (const float*)d_in[15];
    const float* b_g2   = (const float*)d_in[16];
    const float* W_head = (const float*)d_in[17];
    const float* b_head = (const float*)d_in[18];
    float* out = (float*)d_out;

    // workspace layout (floats): ~5MB total
    float* ws   = (float*)d_ws;
    float* X    = ws;                         // NPAD*D1          = 65536
    float* X2T  = X    + NPAD * D1;           // D1*NPAD          = 65536
    float* Ahat = X2T  + D1 * NPAD;           // NPAD*NPAD        = 256
    float* Hp   = Ahat + NPAD * NPAD;         // KSPLIT*NPAD*D2   = 1048576
    float* H    = Hp   + (size_t)KSPLIT * NPAD * D2;  // NPAD*D2  = 131072

    enc_kernel<<<(NPAD * D1) / 256, 256, 0, stream>>>(
        mats, cyls, planes, power, W_mat, b_mat, W_cyl, b_cyl,
        W_pl, b_pl, W_pw, b_pw, X);

    ahat_kernel<<<1, 32, 0, stream>>>(ei, Ahat);

    x2t_kernel<<<(D1 * NPAD) / 256, 256, 0, stream>>>(X, Ahat, X2T);

    // 2048 waves (256 tile-groups x 8 K-chunks), 8 waves per 256-thread block
    gemm_kernel<<<(NWAVE * 32) / 256, 256, 0, stream>>>(X2T, W_g1, Hp);

    reduce_kernel<<<(NPAD * D2) / 256, 256, 0, stream>>>(Hp, b_g1, H);

    final_kernel<<<1, 256, 0, stream>>>(H, Ahat, W_g2, b_g2, W_head, b_head, out);
}